// Bandpass_89472758710535
// MI455X (gfx1250) — compile-verified
//
#include <hip/hip_runtime.h>
#include <math.h>

// ---------------------------------------------------------------------------
// Order-8 Butterworth bandpass filtfilt (scipy-compatible), f64 internals.
// Exact segmented-parallel IIR: per-segment zero-state run (phase A),
// sequential affine compose with A^SEG (phase B, TDM-staged + TDM-pipelined
// operands in LDS), per-segment re-run with correct initial states (phase C).
// ---------------------------------------------------------------------------

#define B_ROWS  32
#define L_LEN   262144
#define PAD     51                    // 3 * max(len(a), len(b)) = 3*17
#define LEXT    (L_LEN + 2*PAD)       // 262246
#define SEG     2048
#define NSEG    129                   // ceil(LEXT / SEG), last segment = 102
#define NSTATE  16
#define NCOEF   17

#define CHUNK   4                     // compose steps per TDM tile
#define NCH     32                    // 128 compose steps / CHUNK
#define CROW    (CHUNK*NSTATE + 1)    // 65 doubles per LDS row (TDM pad)

// workspace layout (units: doubles)
#define OFF_A   0                     // a[17]
#define OFF_B   32                    // b[17]
#define OFF_BP  64                    // bp[16] = b[1:] - a[1:]*b0
#define OFF_ZI  80                    // zi[16]
#define OFF_M   128                   // A^SEG, 16x16 row-major
#define OFF_C   384                   // [32][129][16] zero-state segment ends
#define OFF_Z   (OFF_C + B_ROWS*NSEG*NSTATE)   // [32][129][16] segment inits
#define OFF_Y   (OFF_Z + B_ROWS*NSEG*NSTATE)   // [32][LEXT] pass-1 output (f64)
// total = OFF_Y + 32*262246 = 8,524,352 doubles ~= 65 MiB of d_ws

// ---------------------------------------------------------------------------
// complex f64 helpers (device)
// ---------------------------------------------------------------------------
struct cxd { double re, im; };
__device__ inline cxd cmul(cxd a, cxd b){ return {a.re*b.re - a.im*b.im, a.re*b.im + a.im*b.re}; }
__device__ inline cxd cadd(cxd a, cxd b){ return {a.re+b.re, a.im+b.im}; }
__device__ inline cxd csub(cxd a, cxd b){ return {a.re-b.re, a.im-b.im}; }
__device__ inline cxd cdiv(cxd a, cxd b){
  double d = b.re*b.re + b.im*b.im;
  return {(a.re*b.re + a.im*b.im)/d, (a.im*b.re - a.re*b.im)/d};
}
__device__ inline cxd csqrtd(cxd w){
  double r = sqrt(w.re*w.re + w.im*w.im);
  double u = sqrt(fmax(0.0, (r + w.re)*0.5));
  double v = sqrt(fmax(0.0, (r - w.re)*0.5));
  if (w.im < 0.0) v = -v;
  return {u, v};
}

// ---------------------------------------------------------------------------
// setup: compute b, a, bp, zi and M = A^SEG in f64 on device
// ---------------------------------------------------------------------------
__global__ void bp_setup(double* __restrict__ ws){
  __shared__ double sa[NCOEF];
  if (threadIdx.x == 0){
    const double PI = 3.14159265358979323846264338327950288;
    // Wn = [0.0625, 0.875]; warped = 2*fs*tan(pi*Wn/fs), fs=2
    double w0 = 4.0 * tan(PI * 0.0625 * 0.5);
    double w1 = 4.0 * tan(PI * 0.875  * 0.5);
    double bw = w1 - w0;
    double wo = sqrt(w0 * w1);
    // analog bandpass poles
    cxd pbp[16];
    for (int k = 0; k < 8; k++){
      int m = -7 + 2*k;
      double ang = PI * (double)m / 16.0;
      cxd p   = { -cos(ang), -sin(ang) };       // -exp(i*pi*m/16)
      cxd plp = { p.re*bw*0.5, p.im*bw*0.5 };
      cxd t   = cmul(plp, plp); t.re -= wo*wo;
      cxd s   = csqrtd(t);
      pbp[k]   = cadd(plp, s);
      pbp[k+8] = csub(plp, s);
    }
    // bilinear transform; prod(4 - p_bp) for gain
    cxd pd[16];
    cxd prod = {1.0, 0.0};
    for (int k = 0; k < 16; k++){
      cxd num = { 4.0 + pbp[k].re,  pbp[k].im };
      cxd den = { 4.0 - pbp[k].re, -pbp[k].im };
      pd[k] = cdiv(num, den);
      prod  = cmul(prod, den);
    }
    double bw8 = bw*bw; bw8 *= bw8; bw8 *= bw8;            // bw^8
    cxd kc = cdiv((cxd){65536.0, 0.0}, prod);              // 4^8 / prod
    double kd = bw8 * kc.re;
    // b = kd * coeffs of (x^2 - 1)^8  (zeros: 8 at +1, 8 at -1)
    double binom[9] = {1,8,28,56,70,56,28,8,1};
    double b[NCOEF];
    for (int i = 0; i < NCOEF; i++) b[i] = 0.0;
    for (int u = 0; u <= 8; u++) b[2*u] = kd * ((u & 1) ? -binom[u] : binom[u]);
    // a = real(poly(pd)), descending powers, a[0] = 1
    cxd c[NCOEF];
    for (int i = 0; i < NCOEF; i++) c[i] = (cxd){0.0, 0.0};
    c[0] = (cxd){1.0, 0.0};
    for (int k = 0; k < 16; k++)
      for (int j = k + 1; j >= 1; j--)
        c[j] = csub(c[j], cmul(pd[k], c[j-1]));
    double a[NCOEF];
    for (int i = 0; i < NCOEF; i++) a[i] = c[i].re;
    // bp = b[1:] - a[1:]*b0
    double bp[NSTATE];
    for (int i = 0; i < NSTATE; i++) bp[i] = b[i+1] - a[i+1]*b[0];
    // zi: solve (I - comp^T) zi = bp, 16x16 Gaussian elimination w/ pivoting
    double Mm[NSTATE][NSTATE+1];
    for (int i = 0; i < NSTATE; i++){
      for (int j = 0; j < NSTATE; j++){
        double compT = (j == 0) ? -a[i+1] : ((i == j-1) ? 1.0 : 0.0);
        Mm[i][j] = ((i == j) ? 1.0 : 0.0) - compT;
      }
      Mm[i][NSTATE] = bp[i];
    }
    for (int col = 0; col < NSTATE; col++){
      int piv = col; double best = fabs(Mm[col][col]);
      for (int r = col + 1; r < NSTATE; r++){
        double v = fabs(Mm[r][col]);
        if (v > best){ best = v; piv = r; }
      }
      if (piv != col)
        for (int j = col; j <= NSTATE; j++){
          double t = Mm[col][j]; Mm[col][j] = Mm[piv][j]; Mm[piv][j] = t;
        }
      double inv = 1.0 / Mm[col][col];
      for (int r = col + 1; r < NSTATE; r++){
        double f = Mm[r][col] * inv;
        for (int j = col; j <= NSTATE; j++) Mm[r][j] -= f * Mm[col][j];
      }
    }
    double zi[NSTATE];
    for (int i = NSTATE - 1; i >= 0; i--){
      double s = Mm[i][NSTATE];
      for (int j = i + 1; j < NSTATE; j++) s -= Mm[i][j] * zi[j];
      zi[i] = s / Mm[i][i];
    }
    for (int i = 0; i < NCOEF;  i++){ ws[OFF_A + i] = a[i]; ws[OFF_B + i] = b[i]; }
    for (int i = 0; i < NSTATE; i++){ ws[OFF_BP + i] = bp[i]; ws[OFF_ZI + i] = zi[i]; }
    for (int i = 0; i < NCOEF;  i++) sa[i] = a[i];
  }
  __syncthreads();
  // threads 0..15: column j of M = A^SEG via SEG applications of x=0 step
  int col = threadIdx.x;
  if (col < NSTATE){
    double z[NSTATE];
    #pragma unroll
    for (int i = 0; i < NSTATE; i++) z[i] = (i == col) ? 1.0 : 0.0;
    for (int t = 0; t < SEG; t++){
      double z0 = z[0];
      #pragma unroll
      for (int i = 0; i < NSTATE - 1; i++) z[i] = fma(-sa[i+1], z0, z[i+1]);
      z[NSTATE-1] = -sa[NSTATE] * z0;
    }
    for (int i = 0; i < NSTATE; i++) ws[OFF_M + i*NSTATE + col] = z[i];
  }
}

// ---------------------------------------------------------------------------
// odd-extension sample fetch (pass 1)
// ---------------------------------------------------------------------------
__device__ inline double ext_val(const float* __restrict__ xr, int k){
  if (k < PAD) return 2.0*(double)xr[0] - (double)xr[PAD - k];
  int t = k - PAD;
  if (t < L_LEN) return (double)xr[t];
  int u = t - L_LEN;                                   // 0..PAD-1
  return 2.0*(double)xr[L_LEN-1] - (double)xr[L_LEN-2-u];
}

// ---------------------------------------------------------------------------
// segment runner: phase 0 => zero-state run, emit c_s; phase 1 => real run
// from composed init, emit outputs. pass 0 => forward on ext(x); pass 1 =>
// forward on reversed pass-1 output, final slice written to out (f32).
// ---------------------------------------------------------------------------
__global__ void bp_seg(const float* __restrict__ x, double* __restrict__ ws,
                       float* __restrict__ out, int pass, int phase){
  int w = blockIdx.x * blockDim.x + threadIdx.x;
  if (w >= B_ROWS * NSEG) return;
  int row = w / NSEG, seg = w % NSEG;
  int start = seg * SEG;
  int len = LEXT - start; if (len > SEG) len = SEG;

  double b[NCOEF], a[NCOEF];
  #pragma unroll
  for (int i = 0; i < NCOEF; i++){ b[i] = ws[OFF_B + i]; a[i] = ws[OFF_A + i]; }

  double z[NSTATE];
  if (phase == 0){
    #pragma unroll
    for (int i = 0; i < NSTATE; i++) z[i] = 0.0;
  } else {
    const double* Zi = ws + OFF_Z + (size_t)(row*NSEG + seg)*NSTATE;
    #pragma unroll
    for (int i = 0; i < NSTATE; i++) z[i] = Zi[i];
  }

  const float*  xr   = x  + (size_t)row * L_LEN;
  const double* Yin  = ws + OFF_Y + (size_t)row * LEXT;
  double*       Yout = ws + OFF_Y + (size_t)row * LEXT;
  float*        outr = out + (size_t)row * L_LEN;

  for (int t = start; t < start + len; t++){
    if ((t & 255) == 0){                       // gfx1250 global_prefetch_b8
      if (pass == 0){
        int pf = t - PAD + 1024;
        if (pf >= 0 && pf < L_LEN) __builtin_prefetch((const void*)(xr + pf), 0, 0);
      } else {
        int pf = LEXT - 1 - t - 1024;
        if (pf >= 0) __builtin_prefetch((const void*)(Yin + pf), 0, 0);
      }
    }
    double xt = (pass == 0) ? ext_val(xr, t) : Yin[LEXT - 1 - t];
    double y = fma(b[0], xt, z[0]);
    #pragma unroll
    for (int i = 0; i < NSTATE - 1; i++)
      z[i] = fma(-a[i+1], y, fma(b[i+1], xt, z[i+1]));
    z[NSTATE-1] = fma(-a[NSTATE], y, b[NSTATE] * xt);
    if (phase == 1){
      if (pass == 0){
        Yout[t] = y;
      } else {
        int idx = LEXT - 1 - t - PAD;          // y[::-1][PAD : PAD+L]
        if (idx >= 0 && idx < L_LEN) outr[idx] = (float)y;
      }
    }
  }
  if (phase == 0){
    double* C = ws + OFF_C + (size_t)(row*NSEG + seg)*NSTATE;
    #pragma unroll
    for (int i = 0; i < NSTATE; i++) C[i] = z[i];
  }
}

// ---------------------------------------------------------------------------
// compose: z_{s+1} = M z_s + c_s. Single wave (32 lanes = 32 rows).
// M (A^SEG) is TDM-loaded into LDS; the c_s stream is TDM-pipelined through
// two LDS buffers as [32 x 64]-element f64 tiles with row stride 2064 elems,
// padded by the TDM (+2 DWORDs per 128 DWORDs) to a 65-double LDS row stride
// so lanes hit distinct banks.
// ---------------------------------------------------------------------------
typedef unsigned int v4u __attribute__((ext_vector_type(4)));
typedef int          v8i __attribute__((ext_vector_type(8)));
typedef int          v4i __attribute__((ext_vector_type(4)));

__global__ void bp_compose(const float* __restrict__ x, double* __restrict__ ws, int pass){
  __shared__ double sM[NSTATE * NSTATE];       // 2 KB tile of A^SEG
  __shared__ double sC[2][B_ROWS * CROW];      // 2 x 16.25 KB c_s tiles

#if __has_builtin(__builtin_amdgcn_tensor_load_to_lds) && __has_builtin(__builtin_amdgcn_s_wait_tensorcnt)
  const bool use_tdm = true;
  // D# builder for a 2-D f64 tile (per CDNA5 ISA ch.8)
  auto issueC = [&](int c, int buf){
    unsigned long long ga = (unsigned long long)(ws + OFF_C + (size_t)c * CHUNK * NSTATE);
    unsigned lds_off = (unsigned)(unsigned long long)(size_t)&sC[buf][0];
    v4u g0 = { 1u,                                          // count=1 (valid D#)
               lds_off,                                     // lds_addr
               (unsigned)ga,                                // global_addr[31:0]
               (unsigned)((ga >> 32) & 0x1FFFFFFull) | (2u << 30) };
    v8i g1 = { (int)((3u << 16) | (1u << 20) | (6u << 22) | (1u << 25)),
               //    data_size=8B | pad_en | pad every 128 DW | pad 2 DW
               (int)(64u << 16),        // tensor_dim0[15:0]=64
               (int)(32u << 16),        // tensor_dim0[31:16]=0 | tensor_dim1=32
               (int)(64u << 16),        // tensor_dim1[31:16]=0 | tile_dim0=64
               32,                      // tile_dim1=32 | tile_dim2=0
               (int)(NSEG * NSTATE),    // tensor_dim0_stride = 2064 elements
               0, 0 };
    v4i z4 = {0,0,0,0};
    v8i z8 = {0,0,0,0,0,0,0,0};
    __builtin_amdgcn_tensor_load_to_lds(g0, g1, z4, z4, z8, 0);
  };
  {
    // M tile: 16x16 f64, dense
    unsigned long long ga = (unsigned long long)(ws + OFF_M);
    unsigned lds_off = (unsigned)(unsigned long long)(size_t)&sM[0];
    v4u g0 = { 1u, lds_off, (unsigned)ga,
               (unsigned)((ga >> 32) & 0x1FFFFFFull) | (2u << 30) };
    v8i g1 = { (int)(3u << 16),
               (int)(16u << 16),        // tensor_dim0=16
               (int)(16u << 16),        // tensor_dim1=16
               (int)(16u << 16),        // tile_dim0=16
               16,                      // tile_dim1=16
               16,                      // tensor_dim0_stride=16
               0, 0 };
    v4i z4 = {0,0,0,0};
    v8i z8 = {0,0,0,0,0,0,0,0};
    __builtin_amdgcn_tensor_load_to_lds(g0, g1, z4, z4, z8, 0);
  }
  issueC(0, 0);
  __builtin_amdgcn_s_wait_tensorcnt(0);
  __asm__ volatile("" ::: "memory");            // TDM wrote LDS
#else
  const bool use_tdm = false;
  for (int i = threadIdx.x; i < NSTATE*NSTATE; i += blockDim.x) sM[i] = ws[OFF_M + i];
#endif
  __syncthreads();

  int row = threadIdx.x;
  if (row >= B_ROWS) return;                    // blockDim.x == 32: all valid

  double x0;
  if (pass == 0) x0 = ext_val(x + (size_t)row * L_LEN, 0);          // ext[0]
  else           x0 = ws[OFF_Y + (size_t)row * LEXT + (LEXT - 1)];  // yr[0]

  double z[NSTATE];
  #pragma unroll
  for (int i = 0; i < NSTATE; i++) z[i] = ws[OFF_ZI + i] * x0;

  double*       Zr = ws + OFF_Z + (size_t)row * NSEG * NSTATE;
  const double* Cg = ws + OFF_C + (size_t)row * NSEG * NSTATE;
  #pragma unroll
  for (int i = 0; i < NSTATE; i++) Zr[i] = z[i];

  int buf = 0;
  for (int c = 0; c < NCH; c++){                // 32 chunks x 4 steps = 128
#if __has_builtin(__builtin_amdgcn_tensor_load_to_lds) && __has_builtin(__builtin_amdgcn_s_wait_tensorcnt)
    if (c + 1 < NCH){
      issueC(c + 1, buf ^ 1);                   // prefetch next tile
      __builtin_amdgcn_s_wait_tensorcnt(1);     // oldest (current tile) done
    } else {
      __builtin_amdgcn_s_wait_tensorcnt(0);
    }
    __asm__ volatile("" ::: "memory");
#endif
    const double* cc = &sC[buf][row * CROW];
    for (int k = 0; k < CHUNK; k++){
      int s = c * CHUNK + k;                    // 0..127
      double zn[NSTATE];
      #pragma unroll
      for (int i = 0; i < NSTATE; i++){
        double acc = use_tdm ? cc[k*NSTATE + i] : Cg[s*NSTATE + i];
        #pragma unroll
        for (int j = 0; j < NSTATE; j++) acc = fma(sM[i*NSTATE + j], z[j], acc);
        zn[i] = acc;
      }
      #pragma unroll
      for (int i = 0; i < NSTATE; i++){ z[i] = zn[i]; Zr[(s+1)*NSTATE + i] = zn[i]; }
    }
    buf ^= 1;
  }
}

// ---------------------------------------------------------------------------
// launcher
// ---------------------------------------------------------------------------
extern "C" void kernel_launch(void* const* d_in, const int* in_sizes, int n_in,
                              void* d_out, int out_size, void* d_ws, size_t ws_size,
                              hipStream_t stream){
  (void)in_sizes; (void)n_in; (void)out_size; (void)ws_size;
  const float* x   = (const float*)d_in[0];
  float*       out = (float*)d_out;
  double*      ws  = (double*)d_ws;   // needs ~66 MiB

  bp_setup<<<1, 64, 0, stream>>>(ws);

  const int workers = B_ROWS * NSEG;  // 4128
  const int blk = 128;
  const int grd = (workers + blk - 1) / blk;

  // pass 1: forward filter on odd-extended x -> Y (f64, in ws)
  bp_seg    <<<grd, blk, 0, stream>>>(x, ws, out, 0, 0);
  bp_compose<<<1,   32,  0, stream>>>(x, ws, 0);
  bp_seg    <<<grd, blk, 0, stream>>>(x, ws, out, 0, 1);

  // pass 2: forward filter on reversed Y -> reversed+sliced into out (f32)
  bp_seg    <<<grd, blk, 0, stream>>>(x, ws, out, 1, 0);
  bp_compose<<<1,   32,  0, stream>>>(x, ws, 1);
  bp_seg    <<<grd, blk, 0, stream>>>(x, ws, out, 1, 1);
}